// LossRetina2_18339510354525
// MI455X (gfx1250) — compile-verified
//
#include <hip/hip_runtime.h>
#include <math.h>

#define B 16
#define A 65536
#define G 64
#define NC 20
#define CHW (1 + NC)
#define EPS16 0.0009765625f

// accumulator slots per batch
#define ACC_POS  0
#define ACC_CONF 1
#define ACC_CLS  2
#define ACC_BOX  3
#define ACC_NEGC 4
#define ACC_NEGL 5
#define ACC_STRIDE 8

#define ITEMS 4

typedef float v2f __attribute__((ext_vector_type(2)));
typedef float v8f __attribute__((ext_vector_type(8)));

// ---------------------------------------------------------------------------
// Wave32 sum-reduction via V_WMMA_F32_16X16X4_F32 (fp32 matrix core).
// A(16x4) holds each lane's value twice (lanes0-15 -> K=0,1; lanes16-31 ->
// K=2,3), B = ones. D[m,n] = 2*(v[m]+v[m+16]). Sum the 8 C VGPRs per lane and
// run again: any element of D2 = 4 * total. Requires EXEC all-ones.
// ---------------------------------------------------------------------------
__device__ __forceinline__ float wave_reduce_wmma(float v) {
  v2f a; a[0] = v; a[1] = v;
  v2f ones; ones[0] = 1.0f; ones[1] = 1.0f;
  v8f c = {0.f, 0.f, 0.f, 0.f, 0.f, 0.f, 0.f, 0.f};
  c = __builtin_amdgcn_wmma_f32_16x16x4_f32(false, a, false, ones,
                                            (short)0, c, false, false);
  float t = c[0] + c[1] + c[2] + c[3] + c[4] + c[5] + c[6] + c[7];
  v2f a2; a2[0] = t; a2[1] = t;
  v8f c2 = {0.f, 0.f, 0.f, 0.f, 0.f, 0.f, 0.f, 0.f};
  c2 = __builtin_amdgcn_wmma_f32_16x16x4_f32(false, a2, false, ones,
                                             (short)0, c2, false, false);
  return c2[0] * 0.25f;
}

// BCE with the reference's probability clipping (p in [1e-6, 1-1e-6]).
__device__ __forceinline__ float bce_y1(float z) {  // target = 1
  float p = 1.0f / (1.0f + expf(-z));
  p = fminf(fmaxf(p, 1e-6f), 1.0f - 1e-6f);
  return -logf(p);
}
__device__ __forceinline__ float bce_y0(float z) {  // target = 0
  float p = 1.0f / (1.0f + expf(-z));
  p = fminf(fmaxf(p, 1e-6f), 1.0f - 1e-6f);
  return -logf(1.0f - p);
}

// ---------------------------------------------------------------------------
// K0: zero accumulators + per-gt best-anchor packs (must rerun every call).
// ---------------------------------------------------------------------------
__global__ void k_init(unsigned long long* __restrict__ best,
                       float* __restrict__ acc) {
  int i = blockIdx.x * blockDim.x + threadIdx.x;
  if (i < B * G) best[i] = 0ull;
  if (i < B * ACC_STRIDE) acc[i] = 0.0f;
}

// ---------------------------------------------------------------------------
// K1: IoU match. GT-outer / 4-anchors-inner; per GT the packed (iou_bits<<32 |
// ~anchor) key is max-reduced across the wave with 64-bit shuffles, so only
// lane 0 issues one LDS ds_max_u64 per GT (no same-address serialization).
// GT boxes staged into LDS via gfx1250 async DMA (global_load_async_to_lds).
// ---------------------------------------------------------------------------
__global__ void __launch_bounds__(256) k_match(
    const float* __restrict__ gboxes, const float* __restrict__ ancs,
    float* __restrict__ iou_max, int* __restrict__ idx_gt,
    unsigned long long* __restrict__ best) {
  const int b = blockIdx.y;
  __shared__ float sg[G * 4];
  __shared__ unsigned long long sbest[G];

  if (threadIdx.x < G) {
    // async-copy one 16B gt box per lane into LDS (ASYNCcnt-tracked DMA)
    unsigned lds_off = (unsigned)(size_t)(&sg[0]) + (unsigned)threadIdx.x * 16u;
    unsigned long long gaddr =
        (unsigned long long)(size_t)(gboxes + (size_t)b * G * 4 +
                                     (size_t)threadIdx.x * 4);
    asm volatile("global_load_async_to_lds_b128 %0, %1, off"
                 :: "v"(lds_off), "v"(gaddr)
                 : "memory");
    sbest[threadIdx.x] = 0ull;
  }
  asm volatile("s_wait_asynccnt 0" ::: "memory");
  __syncthreads();

  const int base = blockIdx.x * blockDim.x * ITEMS + threadIdx.x;

  // preload this thread's 4 anchors (ltrb + area), reference-faithful math
  float al[ITEMS], au[ITEMS], ar[ITEMS], ad[ITEMS], areaA[ITEMS];
  float bi[ITEMS];
  int bg[ITEMS];
#pragma unroll
  for (int it = 0; it < ITEMS; ++it) {
    const int a = base + it * 256;
    float cx = ancs[a * 4 + 0], cy = ancs[a * 4 + 1];
    float w  = ancs[a * 4 + 2], h  = ancs[a * 4 + 3];
    al[it] = cx - 0.5f * w; au[it] = cy - 0.5f * h;
    ar[it] = cx + 0.5f * w; ad[it] = cy + 0.5f * h;
    areaA[it] = (ar[it] - al[it]) * (ad[it] - au[it]);
    bi[it] = -1.0f; bg[it] = 0;
  }

  for (int g = 0; g < G; ++g) {
    const float gl = sg[g * 4 + 0], gt = sg[g * 4 + 1];
    const float gr = sg[g * 4 + 2], gb = sg[g * 4 + 3];
    const float areaB = (gr - gl) * (gb - gt);
    unsigned long long mypack = 0ull;
#pragma unroll
    for (int it = 0; it < ITEMS; ++it) {
      const int a = base + it * 256;
      float iw = fminf(ar[it], gr) - fmaxf(al[it], gl);
      float ih = fminf(ad[it], gb) - fmaxf(au[it], gt);
      iw = fmaxf(iw, 0.0f); ih = fmaxf(ih, 0.0f);
      float inter = iw * ih;
      float iou = inter / (areaA[it] + areaB - inter + 1e-12f);
      if (iou > bi[it]) { bi[it] = iou; bg[it] = g; }  // first-max tie-break
      unsigned long long pack =
          (((unsigned long long)__float_as_uint(iou)) << 32) |
          (unsigned long long)(unsigned)(~(unsigned)a);
      if (pack > mypack) mypack = pack;
    }
    // wave32 max-reduction of the packed key (uniform flow, EXEC all-ones)
#pragma unroll
    for (int off = 16; off > 0; off >>= 1) {
      unsigned long long o = __shfl_xor(mypack, off, 32);
      if (o > mypack) mypack = o;
    }
    if ((threadIdx.x & 31) == 0) atomicMax(&sbest[g], mypack);
  }

#pragma unroll
  for (int it = 0; it < ITEMS; ++it) {
    const int a = base + it * 256;
    iou_max[b * A + a] = bi[it];
    idx_gt[b * A + a] = bg[it];
  }
  __syncthreads();
  if (threadIdx.x < G) atomicMax(&best[b * G + threadIdx.x], sbest[threadIdx.x]);
}

// ---------------------------------------------------------------------------
// K2: force-match — each GT's best anchor becomes positive with idx_gt = g.
// Sequential per batch so duplicate best-anchors resolve deterministically.
// ---------------------------------------------------------------------------
__global__ void k_force(const unsigned long long* __restrict__ best,
                        float* __restrict__ iou_max, int* __restrict__ idx_gt) {
  const int b = blockIdx.x;
  if (threadIdx.x == 0) {
    for (int g = 0; g < G; ++g) {
      unsigned long long p = best[b * G + g];
      unsigned a = ~(unsigned)(p & 0xffffffffull);
      iou_max[b * A + a] = 2.0f;
      idx_gt[b * A + a] = g;
    }
  }
}

// ---------------------------------------------------------------------------
// K3: per-anchor losses. Accumulates (pos count, conf-pos, cls, box, neg count)
// with WMMA wave reductions + per-batch global atomics; writes per-anchor
// negative conf loss (or -FLT_MAX sentinel) for OHEM.
// ---------------------------------------------------------------------------
__global__ void __launch_bounds__(256) k_loss(
    const float* __restrict__ ptxywh, const float* __restrict__ pcat,
    const float* __restrict__ gboxes, const int* __restrict__ glabels,
    const float* __restrict__ ancs, const float* __restrict__ iou_max,
    const int* __restrict__ idx_gt, float* __restrict__ negval,
    float* __restrict__ acc) {
  const int b = blockIdx.y;
  const int base = blockIdx.x * blockDim.x * ITEMS + threadIdx.x;

  float s_pos = 0.f, s_conf = 0.f, s_cls = 0.f, s_box = 0.f, s_negc = 0.f;

  for (int it = 0; it < ITEMS; ++it) {
    const int a = base + it * 256;
    const size_t ba = (size_t)b * A + a;
    float im = iou_max[ba];
    bool pos = (im >= 0.5f);
    bool neg = (im < 0.4f);
    float z0 = pcat[ba * CHW];

    float nv = -3.4e38f;
    if (neg) { nv = bce_y0(z0); s_negc += 1.0f; }
    negval[ba] = nv;

    if (pos) {
      s_pos += 1.0f;
      s_conf += bce_y1(z0);

      int g = idx_gt[ba];
      int lab = glabels[b * G + g] - 1;
      for (int c = 0; c < NC; ++c) {
        float z = pcat[ba * CHW + 1 + c];
        s_cls += (c == lab) ? bce_y1(z) : bce_y0(z);
      }
      // box targets
      float gl = gboxes[(b * G + g) * 4 + 0], gt = gboxes[(b * G + g) * 4 + 1];
      float gr = gboxes[(b * G + g) * 4 + 2], gb = gboxes[(b * G + g) * 4 + 3];
      float gx = (gl + gr) * 0.5f, gy = (gt + gb) * 0.5f;
      float gw = gr - gl, gh = gb - gt;
      float axc = ancs[a * 4 + 0], ayc = ancs[a * 4 + 1];
      float aw  = ancs[a * 4 + 2], ah  = ancs[a * 4 + 3];
      float tv[4];
      tv[0] = (gx - axc) / aw * 10.0f;   // / VAR_XY (0.1)
      tv[1] = (gy - ayc) / ah * 10.0f;
      tv[2] = logf(gw / aw) * 5.0f;      // / VAR_WH (0.2)
      tv[3] = logf(gh / ah) * 5.0f;
      for (int i = 0; i < 4; ++i) {
        float d = ptxywh[ba * 4 + i] - tv[i];
        float adv = fabsf(d);
        s_box += (adv < 1.0f) ? 0.5f * d * d : adv - 0.5f;
      }
    }
  }

  // uniform control flow here: EXEC all-ones -> WMMA wave reductions legal
  float r_pos  = wave_reduce_wmma(s_pos);
  float r_conf = wave_reduce_wmma(s_conf);
  float r_cls  = wave_reduce_wmma(s_cls);
  float r_box  = wave_reduce_wmma(s_box);
  float r_negc = wave_reduce_wmma(s_negc);
  if ((threadIdx.x & 31) == 0) {
    atomicAdd(&acc[b * ACC_STRIDE + ACC_POS],  r_pos);
    atomicAdd(&acc[b * ACC_STRIDE + ACC_CONF], r_conf);
    atomicAdd(&acc[b * ACC_STRIDE + ACC_CLS],  r_cls);
    atomicAdd(&acc[b * ACC_STRIDE + ACC_BOX],  r_box);
    atomicAdd(&acc[b * ACC_STRIDE + ACC_NEGC], r_negc);
  }
}

// ---------------------------------------------------------------------------
// K4: OHEM via 4-pass byte-radix select (5 scans total instead of 33).
// Keys = float bits of the negative conf loss (>0 => bit pattern monotonic);
// masked anchors -> key 0, excluded by k <= neg_count. Exact tie handling:
// sum(top-k) = sum(u > kth) + (k - count(u > kth)) * kth.
// ---------------------------------------------------------------------------
__global__ void __launch_bounds__(256) k_ohem(const float* __restrict__ negval,
                                              float* __restrict__ acc) {
  const int b = blockIdx.x, tid = threadIdx.x;
  __shared__ unsigned hist[256];
  __shared__ unsigned s_sel[2];
  __shared__ unsigned s_cnt;
  __shared__ float s_sum;

  int cp = (int)(acc[b * ACC_STRIDE + ACC_POS]  + 0.5f);
  int nc = (int)(acc[b * ACC_STRIDE + ACC_NEGC] + 0.5f);
  int k = (cp > 0) ? 3 * cp : 1;   // num_pos==0 -> rank<3*EPS16 selects top-1
  if (k > nc) k = nc;

  const float* nv = negval + (size_t)b * A;

  if (k > 0) {  // uniform branch (same globals read by all threads)
    unsigned prefix = 0;
    int krem = k;
#pragma unroll
    for (int p = 3; p >= 0; --p) {
      const unsigned shift = (unsigned)p * 8u;
      const unsigned hs = shift + 8u;
      hist[tid] = 0u;
      __syncthreads();
      for (int i = tid; i < A; i += 256) {
        float v = nv[i];
        unsigned u = (v >= 0.f) ? __float_as_uint(v) : 0u;
        // high bytes must match already-chosen prefix (64-bit shift: hs==32 ok)
        if (((unsigned long long)u >> hs) ==
            ((unsigned long long)prefix >> hs)) {
          atomicAdd(&hist[(u >> shift) & 0xFFu], 1u);
        }
      }
      __syncthreads();
      if (tid == 0) {
        unsigned cum = 0, sel = 0;
        for (int bb = 255; bb >= 0; --bb) {
          unsigned c = hist[bb];
          if (cum + c >= (unsigned)krem) { sel = (unsigned)bb; break; }
          cum += c;
        }
        s_sel[0] = sel;
        s_sel[1] = cum;  // count of keys strictly above chosen bucket
      }
      __syncthreads();
      prefix |= s_sel[0] << shift;
      krem -= (int)s_sel[1];
      __syncthreads();
    }

    // final pass: exact count/sum of keys strictly greater than the kth key
    if (tid == 0) { s_cnt = 0u; s_sum = 0.f; }
    __syncthreads();
    unsigned c = 0; float s = 0.f;
    for (int i = tid; i < A; i += 256) {
      float v = nv[i];
      unsigned u = (v >= 0.f) ? __float_as_uint(v) : 0u;
      if (u > prefix) { c++; s += v; }
    }
    atomicAdd(&s_cnt, c);
    atomicAdd(&s_sum, s);
    __syncthreads();
    if (tid == 0) {
      float kth = __uint_as_float(prefix);
      acc[b * ACC_STRIDE + ACC_NEGL] =
          s_sum + (float)(k - (int)s_cnt) * kth;
    }
  } else {
    if (tid == 0) acc[b * ACC_STRIDE + ACC_NEGL] = 0.0f;
  }
}

// ---------------------------------------------------------------------------
// K5: combine -> scalar.
// ---------------------------------------------------------------------------
__global__ void k_final(const float* __restrict__ acc, float* __restrict__ out) {
  if (blockIdx.x == 0 && threadIdx.x == 0) {
    float s = 0.f;
    for (int b = 0; b < B; ++b) {
      float np = fmaxf(acc[b * ACC_STRIDE + ACC_POS], EPS16);
      s += (acc[b * ACC_STRIDE + ACC_CONF] + acc[b * ACC_STRIDE + ACC_NEGL] +
            acc[b * ACC_STRIDE + ACC_CLS]  + acc[b * ACC_STRIDE + ACC_BOX]) / np;
    }
    out[0] = s / (float)B;
  }
}

extern "C" void kernel_launch(void* const* d_in, const int* in_sizes, int n_in,
                              void* d_out, int out_size, void* d_ws, size_t ws_size,
                              hipStream_t stream) {
  const float* ptxywh = (const float*)d_in[0];   // (16,65536,4)
  const float* pcat   = (const float*)d_in[1];   // (16,65536,21)
  const float* gboxes = (const float*)d_in[2];   // (16,64,4)
  const int*   glab   = (const int*)d_in[3];     // (16,64)
  const float* ancs   = (const float*)d_in[4];   // (65536,4)
  float* out = (float*)d_out;

  // workspace carve-up (~12.6 MB)
  float* ws = (float*)d_ws;
  float* iou_max = ws;                              // B*A f32
  int*   idx_gt  = (int*)(ws + (size_t)B * A);      // B*A i32
  float* negval  = ws + 2 * (size_t)B * A;          // B*A f32
  unsigned long long* best =
      (unsigned long long*)(ws + 3 * (size_t)B * A);  // B*G u64 (8B aligned)
  float* acc = (float*)(best + B * G);              // B*8 f32

  k_init<<<(B * G + 255) / 256, 256, 0, stream>>>(best, acc);

  dim3 grid(A / (256 * ITEMS), B);  // (64, 16)
  k_match<<<grid, 256, 0, stream>>>(gboxes, ancs, iou_max, idx_gt, best);
  k_force<<<B, 32, 0, stream>>>(best, iou_max, idx_gt);
  k_loss<<<grid, 256, 0, stream>>>(ptxywh, pcat, gboxes, glab, ancs,
                                   iou_max, idx_gt, negval, acc);
  k_ohem<<<B, 256, 0, stream>>>(negval, acc);
  k_final<<<1, 1, 0, stream>>>(acc, out);
}